// OFPenalty_35656818491466
// MI455X (gfx1250) — compile-verified
//
#include <hip/hip_runtime.h>
#include <hip/hip_bf16.h>

typedef __attribute__((ext_vector_type(2))) float v2f;
typedef __attribute__((ext_vector_type(8))) float v8f;

#define B_BATCH 256
#define C_CH    2048
#define N_DIM   49
#define NPAD    64
#define KC      64

// ---------------------------------------------------------------------------
// Kernel 1: per-batch Gram matrix ATA = X^T X (X = [2048 x 49]) via
// V_WMMA_F32_16X16X4_F32. Output padded to 64x64 (pad cols are zero / unused).
// One workgroup per batch, 8 waves, each wave owns 2 of the 16 output tiles.
// ---------------------------------------------------------------------------
__global__ __launch_bounds__(256) void gram_kernel(const float* __restrict__ x,
                                                   float* __restrict__ gram) {
    __shared__ float xs[KC * NPAD];   // 16 KB: 64 channels x 64 (padded) cols

    const int b    = blockIdx.x;
    const int tid  = threadIdx.x;
    const int lane = tid & 31;
    const int wave = tid >> 5;
    const int cl   = lane & 15;             // column-within-tile for this lane
    const int koff = (lane < 16) ? 0 : 2;   // K offset per ISA A/B VGPR layout

    // zero LDS once; chunk loads only touch cols 0..48 so pad stays zero
    for (int i = tid; i < KC * NPAD; i += 256) xs[i] = 0.0f;

    // tile assignment: wave w -> (mt = w>>1, nt in {2*(w&1), 2*(w&1)+1})
    const int mt  = wave >> 1;
    const int n0a = (wave & 1) * 32;        // = nt0*16
    const int n0b = n0a + 16;               // = nt1*16
    const int m0  = mt * 16;

    v8f acc0 = {};
    v8f acc1 = {};

    const float* xb = x + (size_t)b * C_CH * N_DIM;

    for (int c0 = 0; c0 < C_CH; c0 += KC) {
        __syncthreads();                    // previous chunk fully consumed
        const float* src = xb + c0 * N_DIM; // contiguous 64*49 floats
        for (int i = tid; i < KC * N_DIM; i += 256) {
            int c = i / N_DIM;
            int n = i - c * N_DIM;
            xs[c * NPAD + n] = src[i];
        }
        __syncthreads();

        #pragma unroll
        for (int kk = 0; kk < KC; kk += 4) {
            // lane holds K = kk+koff (x) and K = kk+koff+1 (y) for its column
            const float* rp = &xs[(kk + koff) * NPAD + cl];
            v2f fm, fa, fb;
            fm.x = rp[m0];   fm.y = rp[NPAD + m0];     // A frag: cols m-range
            fa.x = rp[n0a];  fa.y = rp[NPAD + n0a];    // B frag: cols nt0
            fb.x = rp[n0b];  fb.y = rp[NPAD + n0b];    // B frag: cols nt1
            acc0 = __builtin_amdgcn_wmma_f32_16x16x4_f32(
                       false, fm, false, fa, (short)0, acc0, false, false);
            acc1 = __builtin_amdgcn_wmma_f32_16x16x4_f32(
                       false, fm, false, fb, (short)0, acc1, false, false);
        }
    }

    // C/D layout: VGPR r -> (M = m0 + r + (lane<16?0:8), N = n0 + (lane&15))
    float* g = gram + (size_t)b * NPAD * NPAD;
    const int rbase = m0 + ((lane < 16) ? 0 : 8);
    #pragma unroll
    for (int r = 0; r < 8; ++r) {
        g[(rbase + r) * NPAD + n0a + cl] = acc0[r];
        g[(rbase + r) * NPAD + n0b + cl] = acc1[r];
    }
}

// ---------------------------------------------------------------------------
// Kernel 2: per-batch double power iteration + penalty.
// 64 threads/block; thread r holds Gram row r in registers (static indices).
// ---------------------------------------------------------------------------
__device__ __forceinline__ float block_reduce64(float v, float* red, int tid) {
    red[tid] = v;
    __syncthreads();
    #pragma unroll
    for (int off = 32; off > 0; off >>= 1) {
        if (tid < off) red[tid] += red[tid + off];
        __syncthreads();
    }
    float s = red[0];
    __syncthreads();
    return s;
}

__global__ __launch_bounds__(64) void power_kernel(const float* __restrict__ gram,
                                                   const float* __restrict__ x0,
                                                   float* __restrict__ pen) {
    __shared__ float sx[N_DIM];
    __shared__ float red[64];

    const int  b   = blockIdx.x;
    const int  r   = threadIdx.x;
    const bool act = (r < N_DIM);

    float row[N_DIM];
    const float* g = gram + (size_t)b * NPAD * NPAD + r * NPAD;
    #pragma unroll
    for (int j = 0; j < N_DIM; ++j) row[j] = act ? g[j] : 0.0f;

    if (act) sx[r] = x0[b * N_DIM + r];
    __syncthreads();

    // ---- first power iteration: largest eigenvalue ----
    for (int it = 0; it < 9; ++it) {
        float y = 0.0f;
        #pragma unroll
        for (int j = 0; j < N_DIM; ++j) y += row[j] * sx[j];
        float nrm2 = block_reduce64(y * y, red, r);
        float inv  = 1.0f / fmaxf(sqrtf(nrm2), 1e-12f);
        if (act) sx[r] = y * inv;
        __syncthreads();
    }
    float largest;
    {
        float y = 0.0f;
        #pragma unroll
        for (int j = 0; j < N_DIM; ++j) y += row[j] * sx[j];
        float xr  = act ? sx[r] : 0.0f;
        float num = block_reduce64(y * xr, red, r);
        float den = block_reduce64(xr * xr, red, r);
        largest = num / den;
    }

    // ---- shift diagonal: row[r] -= largest (predicated, keeps static idx) ----
    #pragma unroll
    for (int j = 0; j < N_DIM; ++j) row[j] -= (j == r) ? largest : 0.0f;

    // ---- second power iteration on shifted matrix, warm start from x1 ----
    for (int it = 0; it < 9; ++it) {
        float y = 0.0f;
        #pragma unroll
        for (int j = 0; j < N_DIM; ++j) y += row[j] * sx[j];
        float nrm2 = block_reduce64(y * y, red, r);
        float inv  = 1.0f / fmaxf(sqrtf(nrm2), 1e-12f);
        if (act) sx[r] = y * inv;
        __syncthreads();
    }
    float tmp;
    {
        float y = 0.0f;
        #pragma unroll
        for (int j = 0; j < N_DIM; ++j) y += row[j] * sx[j];
        float xr  = act ? sx[r] : 0.0f;
        float num = block_reduce64(y * xr, red, r);
        float den = block_reduce64(xr * xr, red, r);
        tmp = num / den;
    }

    if (r == 0) {
        float smallest = tmp + largest;
        float rat      = largest / smallest - 1.0f;
        pen[b] = rat * rat;   // BETA = 1
    }
}

// ---------------------------------------------------------------------------
// Kernel 3: deterministic sum of 256 penalties / 256
// ---------------------------------------------------------------------------
__global__ __launch_bounds__(256) void sum_kernel(const float* __restrict__ pen,
                                                  float* __restrict__ out) {
    __shared__ float red[256];
    const int t = threadIdx.x;
    red[t] = pen[t];
    __syncthreads();
    #pragma unroll
    for (int off = 128; off > 0; off >>= 1) {
        if (t < off) red[t] += red[t + off];
        __syncthreads();
    }
    if (t == 0) out[0] = red[0] * (1.0f / (float)B_BATCH);
}

extern "C" void kernel_launch(void* const* d_in, const int* in_sizes, int n_in,
                              void* d_out, int out_size, void* d_ws, size_t ws_size,
                              hipStream_t stream) {
    const float* x   = (const float*)d_in[0];   // [256, 2048, 7, 7] f32
    const float* x0  = (const float*)d_in[1];   // [256, 49, 1] f32
    float*       out = (float*)d_out;           // scalar f32

    float* gram = (float*)d_ws;                               // 256*64*64 f32 = 4 MB
    float* pen  = gram + (size_t)B_BATCH * NPAD * NPAD;       // 256 f32

    gram_kernel <<<B_BATCH, 256, 0, stream>>>(x, gram);
    power_kernel<<<B_BATCH,  64, 0, stream>>>(gram, x0, pen);
    sum_kernel  <<<1,       256, 0, stream>>>(pen, out);
}